// StateProbabilitiesTimeDep_22024592293960
// MI455X (gfx1250) — compile-verified
//
#include <hip/hip_runtime.h>

typedef float v2f __attribute__((ext_vector_type(2)));
typedef float v8f __attribute__((ext_vector_type(8)));

namespace {
constexpr int kB = 2048;
constexpr int kT = 256;
constexpr int kND = 8;
constexpr int kD = 16;
constexpr int kTerms = 4;     // Taylor order: error ~ (0.01)^5/120, far below fp32 eps
constexpr int kStride = 18;   // padded row stride (floats): conflict-free on 64 banks
}

__global__ __launch_bounds__(32)
void qprop_kernel(const float* __restrict__ x,
                  const float* __restrict__ h0r, const float* __restrict__ h0i,
                  const float* __restrict__ hdr, const float* __restrict__ hdi,
                  float* __restrict__ out)
{
  __shared__ float Pre[kD * kStride];
  __shared__ float Pim[kD * kStride];
  __shared__ float Tre[kD * kStride];
  __shared__ float Tim[kD * kStride];

  const int lane = threadIdx.x;   // wave32
  const int i = lane & 15;        // A-layout: row M this lane owns; B/C/D: column N
  const int h = lane >> 4;        // half-wave select
  const int b = blockIdx.x;
  const float dt = 1.0f / (float)kT;

  // ---- Preamble: hermitize + dt-scale H0/Hd directly into per-lane A-layout slices.
  // A (16x4 f32) layout per ISA: lane L & L+16 both row M=L&15; K = 4c + v + 2h.
  float hre[1 + kND][8];
  float him[1 + kND][8];
#pragma unroll
  for (int m = 0; m < 1 + kND; ++m) {
    const float* R = (m == 0) ? h0r : (hdr + (size_t)(m - 1) * kD * kD);
    const float* I = (m == 0) ? h0i : (hdi + (size_t)(m - 1) * kD * kD);
#pragma unroll
    for (int p = 0; p < 8; ++p) {
      const int c = p >> 1, v = p & 1;
      const int j = 4 * c + v + 2 * h;
      hre[m][p] = 0.5f * dt * (R[i * kD + j] + R[j * kD + i]);
      him[m][p] = 0.5f * dt * (I[i * kD + j] - I[j * kD + i]);
    }
  }

  // ---- P = identity (held in regs in D-layout, mirrored row-major in LDS)
  v8f Sre, Sim;
#pragma unroll
  for (int r = 0; r < 8; ++r) {
    const int M = r + 8 * h;            // D-layout: VGPR r <-> rows r (lo), r+8 (hi)
    Sre[r] = (i == M) ? 1.0f : 0.0f;
    Sim[r] = 0.0f;
    Pre[M * kStride + i] = Sre[r];
    Pim[M * kStride + i] = Sim[r];
  }
  __syncthreads();

  for (int t = 0; t < kT; ++t) {
    // Drive amplitudes are wave-uniform -> scalar (s_load) path
    const float* xp = x + ((size_t)b * kT + t) * kND;
    float xv[kND];
#pragma unroll
    for (int d = 0; d < kND; ++d) xv[d] = xp[d];

    // Hs = dt*(H0 + sum_d x_d Hd), built in A layout; f32 WMMA has no A-NEG -> pre-negate Im
    float are[8], aim[8], ain[8];
#pragma unroll
    for (int p = 0; p < 8; ++p) {
      float re = hre[0][p], im = him[0][p];
#pragma unroll
      for (int d = 0; d < kND; ++d) {
        re = fmaf(xv[d], hre[1 + d][p], re);
        im = fmaf(xv[d], him[1 + d][p], im);
      }
      are[p] = re; aim[p] = im; ain[p] = -im;
    }

    // B operand := P  (B 4x16 layout: chunk c, VGPR v -> row K = 4c + v + 2h, col = i)
    float bre[8], bim[8];
#pragma unroll
    for (int c = 0; c < 4; ++c)
#pragma unroll
      for (int v = 0; v < 2; ++v) {
        const int row = 4 * c + v + 2 * h;
        bre[c * 2 + v] = Pre[row * kStride + i];
        bim[c * 2 + v] = Pim[row * kStride + i];
      }

    // Taylor: S = P + sum_{k=1..4} T_k,  T_k = (-i/k) * Hs * T_{k-1}
#pragma unroll
    for (int k = 1; k <= kTerms; ++k) {
      v8f mre = {0,0,0,0,0,0,0,0};
      v8f mim = {0,0,0,0,0,0,0,0};
#pragma unroll
      for (int c = 0; c < 4; ++c) {       // chained K-slices: 16x16x16 per real matmul
        v2f aR = {are[2*c], are[2*c+1]};
        v2f aI = {aim[2*c], aim[2*c+1]};
        v2f aN = {ain[2*c], ain[2*c+1]};
        v2f bR = {bre[2*c], bre[2*c+1]};
        v2f bI = {bim[2*c], bim[2*c+1]};
        // M_re = Hre*Tre - Him*Tim ; M_im = Hre*Tim + Him*Tre
        mre = __builtin_amdgcn_wmma_f32_16x16x4_f32(false, aR, false, bR, (short)0, mre, false, false);
        mre = __builtin_amdgcn_wmma_f32_16x16x4_f32(false, aN, false, bI, (short)0, mre, false, false);
        mim = __builtin_amdgcn_wmma_f32_16x16x4_f32(false, aR, false, bI, (short)0, mim, false, false);
        mim = __builtin_amdgcn_wmma_f32_16x16x4_f32(false, aI, false, bR, (short)0, mim, false, false);
      }
      const float inv = 1.0f / (float)k;
      v8f tre, tim;                        // multiply by -i/k : re<->im swap + sign
#pragma unroll
      for (int r = 0; r < 8; ++r) {
        tre[r] =  mim[r] * inv;
        tim[r] = -mre[r] * inv;
        Sre[r] += tre[r];
        Sim[r] += tim[r];
      }
      if (k < kTerms) {
        // D-layout result -> LDS -> reload as next B operand
#pragma unroll
        for (int r = 0; r < 8; ++r) {
          const int M = r + 8 * h;
          Tre[M * kStride + i] = tre[r];
          Tim[M * kStride + i] = tim[r];
        }
        __syncthreads();                   // single-wave WG: lowers to S_NOP, orders LDS
#pragma unroll
        for (int c = 0; c < 4; ++c)
#pragma unroll
          for (int v = 0; v < 2; ++v) {
            const int row = 4 * c + v + 2 * h;
            bre[c * 2 + v] = Tre[row * kStride + i];
            bim[c * 2 + v] = Tim[row * kStride + i];
          }
      }
    }

    // P := S (persist for next step's B-operand load)
#pragma unroll
    for (int r = 0; r < 8; ++r) {
      const int M = r + 8 * h;
      Pre[M * kStride + i] = Sre[r];
      Pim[M * kStride + i] = Sim[r];
    }
    __syncthreads();
  }

  // psi = P e0 (column 0); output |psi|^2
  if (h == 0) {
    const float pr = Pre[i * kStride + 0];
    const float pi = Pim[i * kStride + 0];
    out[(size_t)b * kD + i] = pr * pr + pi * pi;
  }
}

extern "C" void kernel_launch(void* const* d_in, const int* in_sizes, int n_in,
                              void* d_out, int out_size, void* d_ws, size_t ws_size,
                              hipStream_t stream) {
  const float* x   = (const float*)d_in[0];  // [B, T, ND] f32
  const float* h0r = (const float*)d_in[1];  // [D, D]
  const float* h0i = (const float*)d_in[2];  // [D, D]
  const float* hdr = (const float*)d_in[3];  // [ND, D, D]
  const float* hdi = (const float*)d_in[4];  // [ND, D, D]
  float* out = (float*)d_out;                // [B, D]
  qprop_kernel<<<dim3(kB), dim3(32), 0, stream>>>(x, h0r, h0i, hdr, hdi, out);
}